// Profile_74096775790900
// MI455X (gfx1250) — compile-verified
//
#include <hip/hip_runtime.h>
#include <math.h>

typedef __attribute__((ext_vector_type(2))) float v2f;
typedef __attribute__((ext_vector_type(8))) float v8f;

#define PDIM  774            // 129*3*2  (flattened ijk reduction dim)
#define KDIM  258            // 2*(M+1) HMM states
#define NTOT  66564          // 258*258 flattened (l,f)
#define BATCH 32
#define DLAT  20
#define NEG_INF (-1.0e32f)   // -1/EPS in the reference

static __device__ __forceinline__ v8f wmma4(v2f a, v2f b, v8f c) {
  // D = A(16x4 f32) * B(4x16 f32) + C(16x16 f32)
  return __builtin_amdgcn_wmma_f32_16x16x4_f32(
      /*neg_a=*/false, a, /*neg_b=*/false, b,
      /*c_mod=*/(short)0, c, /*reuse_a=*/false, /*reuse_b=*/false);
}

// transition_logits[b, n] = sum_p del[b,p]*uT[p,n] + ins[b,p]*rT[p,n] + NEG_INF*nT[n]
// One wave -> 16 columns x 32 batches. Four independent WMMA accumulator chains
// (U-pass / R-pass x two 16-batch tiles) for ILP; summed at the end.
__global__ __launch_bounds__(256)
void transition_kernel(const float* __restrict__ del_,  // (32,774)
                       const float* __restrict__ ins_,  // (32,774)
                       const float* __restrict__ uT,    // (774,66564)
                       const float* __restrict__ rT,    // (774,66564)
                       const float* __restrict__ nT,    // (66564)
                       float* __restrict__ out)         // (32,66564)
{
  const int lane = threadIdx.x & 31;
  const int wave = threadIdx.x >> 5;
  const int n0   = (blockIdx.x * 8 + wave) * 16;
  const int nc   = n0 + (lane & 15);                 // this lane's column (B/C fragment N)
  const int ncc  = nc < NTOT ? nc : (NTOT - 1);      // clamped for loads
  const int kgrp = (lane >> 4) << 1;                 // K group: 0 (lanes 0-15) or 2 (16-31)
  const int m0   = lane & 15;                        // A row, tile 0 (batches 0..15)
  const int m1   = m0 + 16;                          // A row, tile 1 (batches 16..31)

  const float* dR0 = del_ + m0 * PDIM;
  const float* dR1 = del_ + m1 * PDIM;
  const float* iR0 = ins_ + m0 * PDIM;
  const float* iR1 = ins_ + m1 * PDIM;
  const float* uC  = uT + ncc;
  const float* rC  = rT + ncc;

  v8f accU0 = {0.f,0.f,0.f,0.f,0.f,0.f,0.f,0.f};
  v8f accU1 = accU0;
  v8f accR0 = accU0;
  v8f accR1 = accU0;

  // Full K steps: k0 = 0,4,...,768 (covers k = 0..771; all lanes in range)
  for (int k0 = 0; k0 <= PDIM - 6; k0 += 4) {
    const int ka = k0 + kgrp;        // K for fragment element .x; .y is ka+1
    const int bo = ka * NTOT;        // max 771*66564 < 2^31
    v2f bu, br;
    bu.x = uC[bo];  bu.y = uC[bo + NTOT];
    br.x = rC[bo];  br.y = rC[bo + NTOT];
    v2f ad0 = *(const v2f*)(dR0 + ka);   // aligned: ka even, rows even-length
    v2f ad1 = *(const v2f*)(dR1 + ka);
    v2f ai0 = *(const v2f*)(iR0 + ka);
    v2f ai1 = *(const v2f*)(iR1 + ka);
    accU0 = wmma4(ad0, bu, accU0);   // 4 independent chains per step
    accU1 = wmma4(ad1, bu, accU1);
    accR0 = wmma4(ai0, br, accR0);
    accR1 = wmma4(ai1, br, accR1);
  }

  // Tail: k = 772,773 valid for kgrp==0; kgrp==2 lanes contribute zero (A masked).
  {
    const int   ka  = 772 + kgrp;                  // 772 or 774 (OOB)
    const float msk = (ka < PDIM) ? 1.0f : 0.0f;
    const int   kac = (ka < PDIM) ? ka : (PDIM - 2);
    const int   bo  = kac * NTOT;
    v2f bu, br;
    bu.x = uC[bo];  bu.y = uC[bo + NTOT];
    br.x = rC[bo];  br.y = rC[bo + NTOT];
    v2f ad0, ad1, ai0, ai1;
    ad0.x = dR0[kac] * msk;  ad0.y = dR0[kac + 1] * msk;
    ad1.x = dR1[kac] * msk;  ad1.y = dR1[kac + 1] * msk;
    ai0.x = iR0[kac] * msk;  ai0.y = iR0[kac + 1] * msk;
    ai1.x = iR1[kac] * msk;  ai1.y = iR1[kac + 1] * msk;
    accU0 = wmma4(ad0, bu, accU0);
    accU1 = wmma4(ad1, bu, accU1);
    accR0 = wmma4(ai0, br, accR0);
    accR1 = wmma4(ai1, br, accR1);
  }

  const float nv = NEG_INF * nT[ncc];
  if (nc < NTOT) {
    // C/D layout: lanes 0-15 -> M = r; lanes 16-31 -> M = 8 + r
    const int b0 = (lane >> 4) * 8;
    float* o0 = out + nc;
#pragma unroll
    for (int r = 0; r < 8; ++r) {
      o0[(b0 + r)      * NTOT] = accU0[r] + accR0[r] + nv;  // batches 0..15
      o0[(b0 + r + 16) * NTOT] = accU1[r] + accR1[r] + nv;  // batches 16..31
    }
  }
}

// initial_logits[b,l] = sum_p del[b,p]*u0[p,l] + ins[b,p]*r0[p,l] + NEG_INF*n0[l]
__global__ void initial_kernel(const float* __restrict__ del_,
                               const float* __restrict__ ins_,
                               const float* __restrict__ u0,   // (774,258)
                               const float* __restrict__ r0,   // (774,258)
                               const float* __restrict__ n0,   // (258)
                               float* __restrict__ out)        // (32,258)
{
  int t = blockIdx.x * blockDim.x + threadIdx.x;
  if (t >= BATCH * KDIM) return;
  int b = t / KDIM, l = t - b * KDIM;
  const float* dr = del_ + b * PDIM;
  const float* ir = ins_ + b * PDIM;
  float s = NEG_INF * n0[l];
  for (int p = 0; p < PDIM; ++p)
    s = fmaf(dr[p], u0[p * KDIM + l], fmaf(ir[p], r0[p * KDIM + l], s));
  out[t] = s;
}

// seq[b,k,d] = sum_i pre[b,i,d]*vx[i,k] + insq[b,i,d]*vc[i,k]
__global__ void seq_kernel(const float* __restrict__ pre,   // (32,129,20)
                           const float* __restrict__ isq,   // (32,129,20)
                           const float* __restrict__ vx,    // (129,258)
                           const float* __restrict__ vc,    // (129,258)
                           float* __restrict__ seq)         // (32,258,20)
{
  int t = blockIdx.x * blockDim.x + threadIdx.x;
  if (t >= BATCH * KDIM * DLAT) return;
  int d = t % DLAT;
  int k = (t / DLAT) % KDIM;
  int b = t / (DLAT * KDIM);
  const float* pb = pre + b * 129 * DLAT + d;
  const float* ib = isq + b * 129 * DLAT + d;
  float s = 0.f;
  for (int i = 0; i < 129; ++i)
    s = fmaf(pb[i * DLAT], vx[i * KDIM + k], fmaf(ib[i * DLAT], vc[i * KDIM + k], s));
  seq[t] = s;
}

// obs[b,k,o] = logsumexp_d( seq[b,k,d] + sub[b,d,o] )
__global__ void obs_kernel(const float* __restrict__ seq,   // (32,258,20)
                           const float* __restrict__ sub,   // (32,20,20)
                           float* __restrict__ out)         // (32,258,20)
{
  int t = blockIdx.x * blockDim.x + threadIdx.x;
  if (t >= BATCH * KDIM) return;
  int b = t / KDIM;
  float sv[DLAT];
#pragma unroll
  for (int d = 0; d < DLAT; ++d) sv[d] = seq[t * DLAT + d];
  const float* sb = sub + b * DLAT * DLAT;
  for (int o = 0; o < DLAT; ++o) {
    float mx = -INFINITY;
#pragma unroll
    for (int d = 0; d < DLAT; ++d) mx = fmaxf(mx, sv[d] + sb[d * DLAT + o]);
    float s = 0.f;
#pragma unroll
    for (int d = 0; d < DLAT; ++d) s += expf(sv[d] + sb[d * DLAT + o] - mx);
    out[t * DLAT + o] = mx + logf(s);
  }
}

extern "C" void kernel_launch(void* const* d_in, const int* in_sizes, int n_in,
                              void* d_out, int out_size, void* d_ws, size_t ws_size,
                              hipStream_t stream)
{
  const float* pre  = (const float*)d_in[0];   // precursor_seq_logits (32,129,20)
  const float* isq  = (const float*)d_in[1];   // insert_seq_logits    (32,129,20)
  const float* ins_ = (const float*)d_in[2];   // insert_logits        (32,129,3,2)
  const float* del_ = (const float*)d_in[3];   // delete_logits        (32,129,3,2)
  const float* sub  = (const float*)d_in[4];   // substitute_logits    (32,20,20)
  const float* r0   = (const float*)d_in[5];   // r_transf_0   (129,3,2,258)
  const float* u0   = (const float*)d_in[6];   // u_transf_0   (129,3,2,258)
  const float* n0   = (const float*)d_in[7];   // null_transf_0 (258)
  const float* rT   = (const float*)d_in[8];   // r_transf     (129,3,2,258,258)
  const float* uT   = (const float*)d_in[9];   // u_transf     (129,3,2,258,258)
  const float* nT   = (const float*)d_in[10];  // null_transf  (258,258)
  const float* vx   = (const float*)d_in[11];  // vx_transf    (129,258)
  const float* vc   = (const float*)d_in[12];  // vc_transf    (129,258)

  float* out_init  = (float*)d_out;                 // 32*258
  float* out_trans = out_init + BATCH * KDIM;       // 32*258*258
  float* out_obs   = out_trans + BATCH * NTOT;      // 32*258*20
  float* seq       = (float*)d_ws;                  // 32*258*20 floats (660 KB)

  initial_kernel<<<(BATCH * KDIM + 255) / 256, 256, 0, stream>>>(
      del_, ins_, u0, r0, n0, out_init);

  transition_kernel<<<(NTOT + 127) / 128, 256, 0, stream>>>(
      del_, ins_, uT, rT, nT, out_trans);

  seq_kernel<<<(BATCH * KDIM * DLAT + 255) / 256, 256, 0, stream>>>(
      pre, isq, vx, vc, seq);

  obs_kernel<<<(BATCH * KDIM + 255) / 256, 256, 0, stream>>>(
      seq, sub, out_obs);
}